// QNet_79173427134702
// MI455X (gfx1250) — compile-verified
//
#include <hip/hip_runtime.h>
#include <math.h>

typedef float v2f __attribute__((ext_vector_type(2)));
typedef float v8f __attribute__((ext_vector_type(8)));

__device__ __forceinline__ float shx(float v, int m) { return __shfl_xor(v, m, 32); }

// ---------------------------------------------------------------------------
// Kernel 0: theta is batch-invariant -> precompute sincos(theta/2) once.
// thsc[2t] = cos(theta[t]/2), thsc[2t+1] = sin(theta[t]/2), t in [0,48)
// ---------------------------------------------------------------------------
__global__ void prep_theta_kernel(const float* __restrict__ theta,
                                  float* __restrict__ thsc) {
  int t = (int)threadIdx.x;
  if (t < 48) {
    float s, c;
    sincosf(0.5f * theta[t], &s, &c);
    thsc[2 * t]     = c;
    thsc[2 * t + 1] = s;
  }
}

// ---------------------------------------------------------------------------
// Kernel 1: 6x6 avg-pool as f32 WMMA GEMM (transposed feed).
//   D[bin, batch] = sum_e W[e,bin] * X[batch,e]
// A (16x4) = weights, synthesized in registers (no loads).
// B (4x16) = image data: lane L holds batch L&15; lanes L/L+16 load adjacent
//            8B of the same row -> coalesces to 16B per row, no duplication.
// Rows>=24 / cols>=24 have zero weight -> never touched (reads only 24x24).
// K split into `nchunk` row-block chunks (partials summed in circuit kernel).
// Within a 6-row block binrow is constant; col/6 is compile-time per (u,half).
// ---------------------------------------------------------------------------
__global__ void pool_wmma_kernel(const float* __restrict__ x,
                                 float* __restrict__ xp,
                                 int bsz, int nchunk, int rpc /*rows per chunk*/) {
  int gw   = (int)((blockIdx.x * blockDim.x + threadIdx.x) >> 5);
  int lane = (int)(threadIdx.x & 31);
  int ntiles = bsz >> 4;
  if (gw >= ntiles * nchunk) return;          // whole-wave exit
  int tile = gw / nchunk;
  int ch   = gw - tile * nchunk;
  int m0   = tile * 16;
  int half = lane >> 4;                       // 0: K rows {0,1}; 1: {2,3}
  int n    = lane & 15;                       // A: bin row (M); B: batch (N)
  int r0   = ch * rpc;

  // col/6 for the two K columns each lane half covers (compile-time tables)
  const int CC0[6] = {0, 0, 1, 2, 2, 3};      // cols 4u,4u+1
  const int CC1[6] = {0, 1, 1, 2, 3, 3};      // cols 4u+2,4u+3

  const float* xrow = x + (size_t)(m0 + n) * 784 + 2 * half;
  v8f acc = {0.f, 0.f, 0.f, 0.f, 0.f, 0.f, 0.f, 0.f};

  for (int rb = 0; rb < rpc; rb += 6) {       // 6-row block: binrow constant
    int blk = (r0 + rb) / 6;                  // scalar
    float wv[6];
#pragma unroll
    for (int u = 0; u < 6; ++u) {
      int ccol = half ? CC1[u] : CC0[u];
      wv[u] = (n == blk * 4 + ccol) ? (1.f / 36.f) : 0.f;
    }
#pragma unroll
    for (int rr = 0; rr < 6; ++rr) {
      const float* p = xrow + (size_t)(r0 + rb + rr) * 28;
      float2 d[6];
#pragma unroll
      for (int u = 0; u < 6; ++u)
        d[u] = *(const float2*)(p + 4 * u);   // 8B aligned
#pragma unroll
      for (int u = 0; u < 6; ++u) {
        v2f a, b;
        a.x = wv[u]; a.y = wv[u];
        b.x = d[u].x; b.y = d[u].y;
        acc = __builtin_amdgcn_wmma_f32_16x16x4_f32(false, a, false, b,
                                                    (short)0, acc, false, false);
      }
    }
  }
  // D: VGPR v -> (M=v, N=lane) lo half, (M=v+8, N=lane-16) hi half
  size_t ob = (size_t)ch * bsz * 16 + (size_t)(m0 + n) * 16 + (half ? 8 : 0);
#pragma unroll
  for (int v = 0; v < 8; ++v) xp[ob + v] = acc[v];
}

// ---------------------------------------------------------------------------
// Kernel 2: 8-qubit statevector, one wave per batch element.
// Amp index i[7:0]: wire w <-> bit (7-w). lane = i[7:3] (wires 0..4),
// local l = i[2:0] (wires 5,6,7). 8 complex amps/lane in registers.
// Consecutive 1q gates per wire fused into ONE complex 2x2 matrix.
// ---------------------------------------------------------------------------
__device__ __forceinline__ void getSC(const float* thsc, const float* theta,
                                      int idx, float& c, float& s) {
  if (thsc) { c = thsc[2 * idx]; s = thsc[2 * idx + 1]; }
  else      { sincosf(0.5f * theta[idx], &s, &c); }
}

// m = {m00r,m00i, m01r,m01i, m10r,m10i, m11r,m11i}
template <int W>
__device__ __forceinline__ void applyM(float (&sr)[8], float (&si)[8],
                                       const float (&m)[8], int lane) {
  if constexpr (W >= 5) {
    constexpr int msk = 1 << (7 - W);
#pragma unroll
    for (int l = 0; l < 8; ++l) {
      if ((l & msk) == 0) {
        int l1 = l | msk;
        float a0r = sr[l], a0i = si[l], a1r = sr[l1], a1i = si[l1];
        sr[l]  = m[0]*a0r - m[1]*a0i + m[2]*a1r - m[3]*a1i;
        si[l]  = m[0]*a0i + m[1]*a0r + m[2]*a1i + m[3]*a1r;
        sr[l1] = m[4]*a0r - m[5]*a0i + m[6]*a1r - m[7]*a1i;
        si[l1] = m[4]*a0i + m[5]*a0r + m[6]*a1i + m[7]*a1r;
      }
    }
  } else {
    constexpr int msk = 1 << (4 - W);
    bool hi = (lane & msk) != 0;
    // lo: out = m00*mine + m01*partner ; hi: out = m11*mine + m10*partner
    float car = hi ? m[6] : m[0], cai = hi ? m[7] : m[1];
    float cbr = hi ? m[4] : m[2], cbi = hi ? m[5] : m[3];
#pragma unroll
    for (int l = 0; l < 8; ++l) {
      float pr = shx(sr[l], msk), pi = shx(si[l], msk);
      float r = sr[l], i = si[l];
      sr[l] = car*r - cai*i + cbr*pr - cbi*pi;
      si[l] = car*i + cai*r + cbr*pi + cbi*pr;
    }
  }
}

template <int C, int T>
__device__ __forceinline__ void gCNOT(float (&sr)[8], float (&si)[8], int lane) {
  if constexpr (C >= 5 && T >= 5) {
    constexpr int cm = 1 << (7 - C), tm = 1 << (7 - T);
#pragma unroll
    for (int l = 0; l < 8; ++l) {
      if ((l & cm) && !(l & tm)) {
        int l1 = l | tm;
        float tr = sr[l]; sr[l] = sr[l1]; sr[l1] = tr;
        float ti = si[l]; si[l] = si[l1]; si[l1] = ti;
      }
    }
  } else if constexpr (C >= 5 && T < 5) {
    constexpr int cm = 1 << (7 - C), tm = 1 << (4 - T);
#pragma unroll
    for (int l = 0; l < 8; ++l) {
      if (l & cm) {                       // compile-time per unrolled l
        sr[l] = shx(sr[l], tm);
        si[l] = shx(si[l], tm);
      }
    }
  } else if constexpr (C < 5 && T >= 5) {
    constexpr int cm = 1 << (4 - C), tm = 1 << (7 - T);
    bool act = (lane & cm) != 0;
#pragma unroll
    for (int l = 0; l < 8; ++l) {
      if ((l & tm) == 0) {
        int l1 = l | tm;
        float nr0 = act ? sr[l1] : sr[l];
        float nr1 = act ? sr[l]  : sr[l1];
        float ni0 = act ? si[l1] : si[l];
        float ni1 = act ? si[l]  : si[l1];
        sr[l] = nr0; sr[l1] = nr1; si[l] = ni0; si[l1] = ni1;
      }
    }
  } else {
    constexpr int cm = 1 << (4 - C), tm = 1 << (4 - T);
    bool act = (lane & cm) != 0;
#pragma unroll
    for (int l = 0; l < 8; ++l) {
      float pr = shx(sr[l], tm), pi = shx(si[l], tm);  // all lanes shuffle
      if (act) { sr[l] = pr; si[l] = pi; }
    }
  }
}

template <int J, int K>
__device__ __forceinline__ void cnotIf(float (&sr)[8], float (&si)[8], int lane) {
  if constexpr (J != K) gCNOT<J, K>(sr, si, lane);
}

template <int L, int J>
__device__ __forceinline__ void stepFused(float (&sr)[8], float (&si)[8],
                                          const float (&cx)[16], const float (&sx)[16],
                                          const float* thsc, const float* theta,
                                          const int (&ks)[8], int lane) {
  int k = ks[J];
  if constexpr (L == 0) {
    float ce = cx[2 * J], se = sx[2 * J];        // RY(x0) half-angle
    float cz = cx[2 * J + 1], sz = sx[2 * J + 1]; // RZ(x1) half-angle
    float m[8];
    if (k == 8) {
      float c1, s1, c2, s2, c3, s3;
      getSC(thsc, theta, (0 * 8 + J) * 3 + 0, c1, s1);
      getSC(thsc, theta, (0 * 8 + J) * 3 + 1, c2, s2);
      getSC(thsc, theta, (0 * 8 + J) * 3 + 2, c3, s3);
      // RZ(x1)*RZ(t1) merge: half-angles add
      float cm = cz * c1 - sz * s1, sm = sz * c1 + cz * s1;
      // M1 = RZ(m) * RY(e)
      float a00r =  cm * ce, a00i = -sm * ce;
      float a01r = -cm * se, a01i =  sm * se;
      float a10r =  cm * se, a10i =  sm * se;
      float a11r =  cm * ce, a11i =  sm * ce;
      // M2 = RY(t2) * M1
      float b00r = c2*a00r - s2*a10r, b00i = c2*a00i - s2*a10i;
      float b01r = c2*a01r - s2*a11r, b01i = c2*a01i - s2*a11i;
      float b10r = s2*a00r + c2*a10r, b10i = s2*a00i + c2*a10i;
      float b11r = s2*a01r + c2*a11r, b11i = s2*a01i + c2*a11i;
      // M = RZ(t3) * M2 : row0 *= (c3 - i s3), row1 *= (c3 + i s3)
      m[0] = b00r*c3 + b00i*s3;  m[1] = b00i*c3 - b00r*s3;
      m[2] = b01r*c3 + b01i*s3;  m[3] = b01i*c3 - b01r*s3;
      m[4] = b10r*c3 - b10i*s3;  m[5] = b10i*c3 + b10r*s3;
      m[6] = b11r*c3 - b11i*s3;  m[7] = b11i*c3 + b11r*s3;
    } else {
      // M = RZ(x1) * RY(x0)
      m[0] =  cz * ce;  m[1] = -sz * ce;
      m[2] = -cz * se;  m[3] =  sz * se;
      m[4] =  cz * se;  m[5] =  sz * se;
      m[6] =  cz * ce;  m[7] =  sz * ce;
    }
    applyM<J>(sr, si, m, lane);
  } else {
    if (k == 8) {
      float c1, s1, c2, s2, c3, s3;
      getSC(thsc, theta, (1 * 8 + J) * 3 + 0, c1, s1);
      getSC(thsc, theta, (1 * 8 + J) * 3 + 1, c2, s2);
      getSC(thsc, theta, (1 * 8 + J) * 3 + 2, c3, s3);
      // A = RY(t2)*RZ(t1)
      float a00r =  c2 * c1, a00i = -c2 * s1;
      float a01r = -s2 * c1, a01i = -s2 * s1;
      float a10r =  s2 * c1, a10i = -s2 * s1;
      float a11r =  c2 * c1, a11i =  c2 * s1;
      float m[8];
      m[0] = a00r*c3 + a00i*s3;  m[1] = a00i*c3 - a00r*s3;
      m[2] = a01r*c3 + a01i*s3;  m[3] = a01i*c3 - a01r*s3;
      m[4] = a10r*c3 - a10i*s3;  m[5] = a10i*c3 + a10r*s3;
      m[6] = a11r*c3 - a11i*s3;  m[7] = a11i*c3 + a11r*s3;
      applyM<J>(sr, si, m, lane);
    }
  }
  if (k < 8) {
    switch (k) {                         // wave-uniform scalar branch
      case 0: cnotIf<J, 0>(sr, si, lane); break;
      case 1: cnotIf<J, 1>(sr, si, lane); break;
      case 2: cnotIf<J, 2>(sr, si, lane); break;
      case 3: cnotIf<J, 3>(sr, si, lane); break;
      case 4: cnotIf<J, 4>(sr, si, lane); break;
      case 5: cnotIf<J, 5>(sr, si, lane); break;
      case 6: cnotIf<J, 6>(sr, si, lane); break;
      case 7: cnotIf<J, 7>(sr, si, lane); break;
      default: break;
    }
  }
}

__global__ void circuit_kernel(const float* __restrict__ img,
                               const float* __restrict__ xp, int nchunk,
                               const float* __restrict__ thsc,
                               const float* __restrict__ theta,
                               const int* __restrict__ hw,
                               float* __restrict__ out, int bsz) {
  int wave = (int)((blockIdx.x * blockDim.x + threadIdx.x) >> 5);
  int lane = (int)(threadIdx.x & 31);
  if (wave >= bsz) return;               // whole-wave exit

  // pooled values: lanes 0..15 hold one each (sum chunk partials)
  float pv = 0.f;
  if (xp) {
    if (lane < 16) {
      size_t o = (size_t)wave * 16 + lane;
      size_t stride = (size_t)bsz * 16;
#pragma unroll
      for (int ch = 0; ch < 4; ++ch)
        if (ch < nchunk) pv += xp[(size_t)ch * stride + o];
    }
  } else if (lane < 16) {
    const float* ib = img + (size_t)wave * 784;
    int by = lane >> 2, bx = lane & 3;
    float ssum = 0.f;
#pragma unroll
    for (int r = 0; r < 6; ++r)
#pragma unroll
      for (int c = 0; c < 6; ++c)
        ssum += ib[(6 * by + r) * 28 + 6 * bx + c];
    pv = ssum * (1.f / 36.f);
  }

  // lane-parallel encoder sincos: each lane does ONE sincos, then broadcast
  float es, ec;
  sincosf(0.5f * pv, &es, &ec);
  float cx[16], sx[16];
#pragma unroll
  for (int t = 0; t < 16; ++t) {
    cx[t] = __shfl(ec, t, 32);
    sx[t] = __shfl(es, t, 32);
  }

  // argmax(hardwts) per wire (first max, like jnp.argmax)
  int ks[8];
#pragma unroll
  for (int j = 0; j < 8; ++j) {
    int best = hw[j * 9], arg = 0;
#pragma unroll
    for (int k = 1; k < 9; ++k) {
      int v = hw[j * 9 + k];
      if (v > best) { best = v; arg = k; }
    }
    ks[j] = arg;
  }

  // |0...0>
  float sr[8], si[8];
#pragma unroll
  for (int l = 0; l < 8; ++l) { sr[l] = 0.f; si[l] = 0.f; }
  if (lane == 0) sr[0] = 1.f;

  stepFused<0, 0>(sr, si, cx, sx, thsc, theta, ks, lane);
  stepFused<0, 1>(sr, si, cx, sx, thsc, theta, ks, lane);
  stepFused<0, 2>(sr, si, cx, sx, thsc, theta, ks, lane);
  stepFused<0, 3>(sr, si, cx, sx, thsc, theta, ks, lane);
  stepFused<0, 4>(sr, si, cx, sx, thsc, theta, ks, lane);
  stepFused<0, 5>(sr, si, cx, sx, thsc, theta, ks, lane);
  stepFused<0, 6>(sr, si, cx, sx, thsc, theta, ks, lane);
  stepFused<0, 7>(sr, si, cx, sx, thsc, theta, ks, lane);
  stepFused<1, 0>(sr, si, cx, sx, thsc, theta, ks, lane);
  stepFused<1, 1>(sr, si, cx, sx, thsc, theta, ks, lane);
  stepFused<1, 2>(sr, si, cx, sx, thsc, theta, ks, lane);
  stepFused<1, 3>(sr, si, cx, sx, thsc, theta, ks, lane);
  stepFused<1, 4>(sr, si, cx, sx, thsc, theta, ks, lane);
  stepFused<1, 5>(sr, si, cx, sx, thsc, theta, ks, lane);
  stepFused<1, 6>(sr, si, cx, sx, thsc, theta, ks, lane);
  stepFused<1, 7>(sr, si, cx, sx, thsc, theta, ks, lane);

  // <Z_w> = sum_i |amp_i|^2 * (+1 if bit_w==0 else -1)
  float p[8], tot = 0.f;
#pragma unroll
  for (int l = 0; l < 8; ++l) { p[l] = sr[l]*sr[l] + si[l]*si[l]; tot += p[l]; }
  float z[8];
#pragma unroll
  for (int w = 0; w < 5; ++w)
    z[w] = ((lane >> (4 - w)) & 1) ? -tot : tot;
#pragma unroll
  for (int w = 5; w < 8; ++w) {
    float a = 0.f;
#pragma unroll
    for (int l = 0; l < 8; ++l)
      a += ((l >> (7 - w)) & 1) ? -p[l] : p[l];
    z[w] = a;
  }
#pragma unroll
  for (int w = 0; w < 8; ++w)
#pragma unroll
    for (int off = 16; off >= 1; off >>= 1)
      z[w] += shx(z[w], off);            // all lanes end with full sums

  // log_softmax
  float mx = z[0];
#pragma unroll
  for (int w = 1; w < 8; ++w) mx = fmaxf(mx, z[w]);
  float se = 0.f;
#pragma unroll
  for (int w = 0; w < 8; ++w) se += expf(z[w] - mx);
  float ls = logf(se);
  float ov = 0.f;                        // avoid runtime-indexed array
#pragma unroll
  for (int w = 0; w < 8; ++w)
    if (lane == w) ov = z[w];
  if (lane < 8)
    out[(size_t)wave * 8 + lane] = ov - mx - ls;
}

// ---------------------------------------------------------------------------
extern "C" void kernel_launch(void* const* d_in, const int* in_sizes, int n_in,
                              void* d_out, int out_size, void* d_ws, size_t ws_size,
                              hipStream_t stream) {
  (void)n_in; (void)out_size;
  const float* x     = (const float*)d_in[0];   // [bsz,1,28,28] f32
  const float* theta = (const float*)d_in[1];   // [2,8,3] f32
  const int*   hw    = (const int*)d_in[2];     // [8,9] i32
  float* out = (float*)d_out;
  int bsz = in_sizes[0] / 784;

  const size_t thscBytes = 96 * sizeof(float);
  int nchunk = 0;
  if (bsz % 16 == 0) {
    if      (ws_size >= (size_t)4 * bsz * 16 * sizeof(float) + thscBytes) nchunk = 4;
    else if (ws_size >= (size_t)1 * bsz * 16 * sizeof(float) + thscBytes) nchunk = 1;
  }

  const float* xp = nullptr;
  const float* thsc = nullptr;
  if (nchunk) {
    float* wsF = (float*)d_ws;
    float* thscW = wsF + (size_t)nchunk * bsz * 16;
    prep_theta_kernel<<<1, 64, 0, stream>>>(theta, thscW);
    int ntiles = bsz / 16;
    int waves = ntiles * nchunk;
    int threads = 256;                          // 8 waves / block
    int blocks = (waves * 32 + threads - 1) / threads;
    pool_wmma_kernel<<<blocks, threads, 0, stream>>>(x, wsF, bsz, nchunk,
                                                     24 / nchunk);
    xp = wsF;
    thsc = thscW;
  }
  int threads2 = 256;                           // 8 waves / block
  int blocks2 = (bsz * 32 + threads2 - 1) / threads2;
  circuit_kernel<<<blocks2, threads2, 0, stream>>>(x, xp, nchunk, thsc, theta,
                                                   hw, out, bsz);
}